// ClusterLoss_54374285967845
// MI455X (gfx1250) — compile-verified
//
#include <hip/hip_runtime.h>

// ---------------------------------------------------------------------------
// ClusterLoss for MI455X (gfx1250, wave32).
//
// loss = ||W||^2 + 0.5*(between + within)/bs
//   within  = sumsq - Sc,            Sc = sum_g ||s_g||^2 / m_g
//   between = Sc - wsum2 / n,        wsum2 = || sum over all columns of W ||^2
// where s_g = per-group column sums (S[1024 x 128]), m_g = group counts.
//
// Pass 1 (memory-bound, ~8.6us floor @ 23.3 TB/s): gid tile staged once per
//   block via GLOBAL_LOAD_ASYNC_TO_LDS_B128 (ASYNCcnt DMA, overlapped with
//   LDS accumulator zeroing), W streamed with b128 loads, squares in VGPRs,
//   group scatter into LDS (ds_add_f32), fp32 global-atomic flush.
// Pass 2 (WMMA): Q_g = diag(S^T S) via v_wmma_f32_16x16x4_f32 (8 diagonal
//   16x16 tiles, K=1024). Per the ISA f32 A(16x4)/B(4x16) lane layouts the
//   two operands are identical register pairs for the symmetric product.
// ---------------------------------------------------------------------------

typedef __attribute__((ext_vector_type(2))) float v2f;
typedef __attribute__((ext_vector_type(8))) float v8f;
typedef __attribute__((ext_vector_type(4))) float f4;
typedef __attribute__((ext_vector_type(4))) int   i4;

#define G_GROUPS 128
#define ROWS     16
#define COLS     1024
#define THREADS  256

#if __has_builtin(__builtin_amdgcn_global_load_async_to_lds_b128) && \
    __has_builtin(__builtin_amdgcn_s_wait_asynccnt)
#define HAS_ASYNC_LDS 1
#else
#define HAS_ASYNC_LDS 0
#endif

// ---------------------------------------------------------------- zero ws
__global__ void k_zero(float* __restrict__ ws, int n) {
  int i = blockIdx.x * blockDim.x + threadIdx.x;
  if (i < n) ws[i] = 0.0f;
}

// ------------------------------------------------- pass 1: stream W once
__global__ void __launch_bounds__(THREADS)
k_pass(const float* __restrict__ W, const int* __restrict__ gid,
       float* __restrict__ S, float* __restrict__ counts,
       float* __restrict__ scal, int D, int N) {
  __shared__ float acc[ROWS * G_GROUPS];   // [row 0..15][group 0..127]
  __shared__ float cnt[G_GROUPS];
  __shared__ float red[THREADS];
  __shared__ int   sgid[COLS];             // staged group ids for this block

  const int t = threadIdx.x;
  const int colbase = blockIdx.x * COLS;

  // ---- stage gid tile into LDS (one b128 per thread = 4KB per block) ----
  {
    const int k   = t * 4;
    int       src = colbase + k;
    if (src > N - 4) src = (N >= 4) ? (N - 4) : 0;   // stay in-bounds; tail
                                                     // slots never consumed
#if HAS_ASYNC_LDS
    typedef __attribute__((address_space(1))) int g_int;   // global scalar
    typedef __attribute__((address_space(3))) int l_int;   // lds scalar
    typedef __attribute__((address_space(1))) i4  g_v4;    // global v4i
    typedef __attribute__((address_space(3))) i4  l_v4;    // lds v4i
    __builtin_amdgcn_global_load_async_to_lds_b128(
        (g_v4*)(g_int*)(gid + src), (l_v4*)(l_int*)&sgid[k], 0, 0);
#else
    *(i4*)&sgid[k] = *(const i4*)(gid + src);
#endif
  }

  // overlap DMA with LDS accumulator init
  for (int i = t; i < ROWS * G_GROUPS; i += THREADS) acc[i] = 0.0f;
  if (t < G_GROUPS) cnt[t] = 0.0f;

#if HAS_ASYNC_LDS
  __builtin_amdgcn_s_wait_asynccnt(0);
#endif
  __syncthreads();

  const int r = t >> 4;           // 16 rows per block
  const int c = t & 15;           // 16 threads per row
  const int d = blockIdx.y * ROWS + r;
  const float* Wrow = W + (size_t)d * (size_t)N;
  const bool do_count = (blockIdx.y == 0) && (r == 0);   // count each col once

  float ss = 0.0f;

  for (int it = 0; it < 16; ++it) {
    const int cl  = (it * 16 + c) * 4;      // column offset within block tile
    const int col = colbase + cl;           // monotone in it for fixed c
    if (col >= N) break;
    if (col + 3 < N) {
      __builtin_prefetch(Wrow + col + 256, 0, 0);   // 1KB-ahead hint
      f4 w = *(const f4*)(Wrow + col);
      i4 g = *(const i4*)&sgid[cl];                 // ds_load_b128
      const int g0 = g.x & 127, g1 = g.y & 127, g2 = g.z & 127, g3 = g.w & 127;
      ss += w.x * w.x + w.y * w.y + w.z * w.z + w.w * w.w;
      atomicAdd(&acc[r * G_GROUPS + g0], w.x);
      atomicAdd(&acc[r * G_GROUPS + g1], w.y);
      atomicAdd(&acc[r * G_GROUPS + g2], w.z);
      atomicAdd(&acc[r * G_GROUPS + g3], w.w);
      if (do_count) {
        atomicAdd(&cnt[g0], 1.0f); atomicAdd(&cnt[g1], 1.0f);
        atomicAdd(&cnt[g2], 1.0f); atomicAdd(&cnt[g3], 1.0f);
      }
    } else {
      for (int j = 0; j < 4 && (col + j) < N; ++j) {   // tail: gid from global
        const float w = Wrow[col + j];
        const int   g = gid[col + j] & 127;
        ss += w * w;
        atomicAdd(&acc[r * G_GROUPS + g], w);
        if (do_count) atomicAdd(&cnt[g], 1.0f);
      }
    }
  }
  __syncthreads();

  // flush group accumulators (fp32 global atomics, L2-resident)
  for (int i = t; i < ROWS * G_GROUPS; i += THREADS) {
    const int dl = i >> 7, g = i & 127;
    unsafeAtomicAdd(&S[(size_t)(blockIdx.y * ROWS + dl) * G_GROUPS + g], acc[i]);
  }
  if (blockIdx.y == 0 && t < G_GROUPS) unsafeAtomicAdd(&counts[t], cnt[t]);

  // block-reduce sum of squares
  red[t] = ss;
  __syncthreads();
  for (int s = THREADS / 2; s > 0; s >>= 1) {
    if (t < s) red[t] += red[t + s];
    __syncthreads();
  }
  if (t == 0) unsafeAtomicAdd(&scal[0], red[0]);
}

// ------------------- pass 2: Q_g = diag(S^T S) via v_wmma_f32_16x16x4_f32
__global__ void __launch_bounds__(THREADS)
k_gram(const float* __restrict__ S, float* __restrict__ Q,
       float* __restrict__ scal, int D) {
  const int t    = threadIdx.x;
  const int lane = t & 31;
  const int w    = t >> 5;                 // wave 0..7 -> diagonal tile
  const int dbase = blockIdx.x * 64;       // this block's K-slice (64 of D)
  const int g0   = w * 16;
  const int gcol = g0 + (lane & 15);       // A: M = group, B: N = group
  const int koff = (lane < 16) ? 0 : 2;    // f32 A 16x4 lane layout: K split

  v8f cacc = {};
  for (int step = 0; step < 16; ++step) {
    const int d0 = dbase + step * 4 + koff;
    v2f a;
    a.x = S[(size_t)d0       * G_GROUPS + gcol];   // A[M][K] = S[d][g]
    a.y = S[(size_t)(d0 + 1) * G_GROUPS + gcol];   // B[K][N] = same regs
    cacc = __builtin_amdgcn_wmma_f32_16x16x4_f32(
        false, a, false, a, (short)0, cacc, false, false);
  }

  // C layout: VGPR v, lanes 0-15 -> (M=v, N=lane); lanes 16-31 -> (M=v+8, N=lane-16)
#pragma unroll
  for (int v = 0; v < 8; ++v) {
    const int M = (lane < 16) ? v : v + 8;
    const int Ncol = (lane < 16) ? lane : lane - 16;
    if (M == Ncol) unsafeAtomicAdd(&Q[g0 + M], cacc[v]);
  }

  // partial ||rowsum||^2 over this block's 64 rows of S
  __shared__ float red[64];
  if (t < 64) {
    const int d = dbase + t;
    float rs = 0.0f;
    for (int g = 0; g < G_GROUPS; ++g) rs += S[(size_t)d * G_GROUPS + g];
    red[t] = rs * rs;
  }
  __syncthreads();
  for (int s = 32; s > 0; s >>= 1) {
    if (t < s) red[t] += red[t + s];
    __syncthreads();
  }
  if (t == 0) unsafeAtomicAdd(&scal[1], red[0]);
}

// ------------------------------------------------------------- final scalar
__global__ void __launch_bounds__(G_GROUPS)
k_final(const float* __restrict__ counts, const float* __restrict__ Q,
        const float* __restrict__ scal, const int* __restrict__ bs_ptr,
        float* __restrict__ out, int N) {
  __shared__ float red[G_GROUPS];
  const int t = threadIdx.x;
  const float m = counts[t];
  red[t] = (m > 0.5f) ? (Q[t] / m) : 0.0f;   // m_g * ||mu_g||^2
  __syncthreads();
  for (int s = G_GROUPS / 2; s > 0; s >>= 1) {
    if (t < s) red[t] += red[t + s];
    __syncthreads();
  }
  if (t == 0) {
    const float Sc    = red[0];
    const float sumsq = scal[0];
    const float wsum2 = scal[1];
    const float n     = (float)N;
    const float bs    = (float)bs_ptr[0];
    const float between = Sc - wsum2 / n;     // = Sc - n*||w_bar||^2
    const float within  = sumsq - Sc;         // ANOVA identity
    out[0] = sumsq * 1.0f + (between * 0.5f + within * 0.5f) / bs;
  }
}

// ---------------------------------------------------------------------------
extern "C" void kernel_launch(void* const* d_in, const int* in_sizes, int n_in,
                              void* d_out, int out_size, void* d_ws, size_t ws_size,
                              hipStream_t stream) {
  const float* W    = (const float*)d_in[0];
  const int*   gids = (const int*)d_in[1];
  const int*   bs   = (const int*)d_in[2];
  const int N = in_sizes[1];
  const int D = in_sizes[0] / N;              // 1024

  float* ws     = (float*)d_ws;
  float* S      = ws;                          // [D][128]
  float* counts = ws + (size_t)D * G_GROUPS;   // [128]
  float* Q      = counts + G_GROUPS;           // [128]
  float* scal   = Q + G_GROUPS;                // [0]=sumsq [1]=wsum2

  const int zn = D * G_GROUPS + 2 * G_GROUPS + 2;
  k_zero<<<(zn + 255) / 256, 256, 0, stream>>>(ws, zn);

  dim3 grid((N + COLS - 1) / COLS, D / ROWS);
  k_pass<<<grid, THREADS, 0, stream>>>(W, gids, S, counts, scal, D, N);

  k_gram<<<D / 64, THREADS, 0, stream>>>(S, Q, scal, D);

  k_final<<<1, G_GROUPS, 0, stream>>>(counts, Q, scal, bs, (float*)d_out, N);
}